// MultiScaleRouter_17901423690026
// MI455X (gfx1250) — compile-verified
//
#include <hip/hip_runtime.h>
#include <math.h>

#define B_   16
#define T_   512
#define D_   512
#define KSEL 4
#define M_   8
#define NKS  6

typedef float v2f __attribute__((ext_vector_type(2)));
typedef float v8f __attribute__((ext_vector_type(8)));

#define PI2_OVER_T (6.283185307179586f / 512.0f)

// ---------------------------------------------------------------------------
// K0: DFT coefficient matrix Wall[512][512]:
//   rows 0..255  : cos(2*pi*f*t/512)      (Re part of rfft)
//   rows 256..511: -sin(2*pi*f*t/512)     (Im part of rfft)
// Angle reduced exactly via (f*t) mod 512.
// ---------------------------------------------------------------------------
__global__ void build_coeffs_kernel(float* __restrict__ Wall) {
    int idx = blockIdx.x * blockDim.x + threadIdx.x;   // 0 .. 512*512-1
    int f = idx >> 9;
    int t = idx & 511;
    int fr = f & 255;
    float ang = (float)((fr * t) & 511) * PI2_OVER_T;
    Wall[idx] = (f < 256) ? cosf(ang) : -sinf(ang);
}

// ---------------------------------------------------------------------------
// K1: batched DFT GEMM via V_WMMA_F32_16X16X4_F32.
//   Xall[b] (512x512) = Wall (512x512) @ x[b] (512x512, t-major rows)
// Each wave computes a 64x64 tile of C (4x4 WMMA tiles, 16 v8f accumulators).
// blockDim = 128 (4 waves); grid = (16 blocks/batch, 16 batches).
// Fragment layouts per CDNA5 ISA 7.12.2 (A: 2 VGPR/lane {K=2h,2h+1} at row m;
// B assumed row-striped: VGPR0=row K=2h, VGPR1=row K=2h+1 at col n;
// C: VGPR v -> row v + 8*half, col = lane&15).
// ---------------------------------------------------------------------------
__global__ __launch_bounds__(128) void dft_gemm_kernel(
        const float* __restrict__ Wall,
        const float* __restrict__ X,
        float* __restrict__ Xall) {
    const int b    = blockIdx.y;
    const int wave = threadIdx.x >> 5;
    const int lane = threadIdx.x & 31;
    const int tile = blockIdx.x * 4 + wave;     // 0..63
    const int tm   = (tile >> 3) * 64;
    const int tn   = (tile & 7) * 64;
    const int m    = lane & 15;
    const int half = lane >> 4;

    const float* __restrict__ Bx = X + (size_t)b * T_ * D_;
    float* __restrict__       Cp = Xall + (size_t)b * 512 * D_;

    v8f acc[4][4];
#pragma unroll
    for (int mi = 0; mi < 4; ++mi)
#pragma unroll
        for (int ni = 0; ni < 4; ++ni)
            acc[mi][ni] = (v8f){0.f, 0.f, 0.f, 0.f, 0.f, 0.f, 0.f, 0.f};

#pragma unroll 1
    for (int kk = 0; kk < 512; kk += 4) {
        const int krow = kk + 2 * half;
        v2f af[4];
        v2f bf[4];
#pragma unroll
        for (int mi = 0; mi < 4; ++mi)
            af[mi] = *(const v2f*)(Wall + (size_t)(tm + mi * 16 + m) * 512 + krow);
#pragma unroll
        for (int ni = 0; ni < 4; ++ni) {
            const int col = tn + ni * 16 + m;
            bf[ni].x = Bx[(size_t)krow * 512 + col];
            bf[ni].y = Bx[(size_t)(krow + 1) * 512 + col];
        }
#pragma unroll
        for (int mi = 0; mi < 4; ++mi)
#pragma unroll
            for (int ni = 0; ni < 4; ++ni)
                acc[mi][ni] = __builtin_amdgcn_wmma_f32_16x16x4_f32(
                    false, af[mi], false, bf[ni], (short)0, acc[mi][ni],
                    false, false);
    }

#pragma unroll
    for (int mi = 0; mi < 4; ++mi)
#pragma unroll
        for (int ni = 0; ni < 4; ++ni) {
            const int col = tn + ni * 16 + m;
#pragma unroll
            for (int v = 0; v < 8; ++v) {
                const int row = tm + mi * 16 + half * 8 + v;
                Cp[(size_t)row * 512 + col] = acc[mi][ni][v];
            }
        }
}

// ---------------------------------------------------------------------------
// K2: per-(b,d) series: top-4 |Xf| bins (f = 1..255), store 2|X|/T, phase, f.
// Strict-> insertion keeps lowest index on ties (lax.top_k semantics).
// ---------------------------------------------------------------------------
__global__ void select_topk_kernel(const float* __restrict__ Xall,
                                   float* __restrict__ pA,
                                   float* __restrict__ pPhi,
                                   int* __restrict__ pF) {
    int idx = blockIdx.x * blockDim.x + threadIdx.x;   // 0..8191
    int b = idx >> 9;
    int d = idx & 511;
    const float* Re = Xall + (size_t)b * 512 * 512 + d;
    const float* Im = Re + 256 * 512;

    float bv0 = -1.f, bv1 = -1.f, bv2 = -1.f, bv3 = -1.f;
    int   bi0 = 1, bi1 = 1, bi2 = 1, bi3 = 1;
    for (int f = 1; f < 256; ++f) {
        float re = Re[(size_t)f * 512];
        float im = Im[(size_t)f * 512];
        float a2 = re * re + im * im;
        if (a2 > bv0)      { bv3=bv2; bi3=bi2; bv2=bv1; bi2=bi1; bv1=bv0; bi1=bi0; bv0=a2; bi0=f; }
        else if (a2 > bv1) { bv3=bv2; bi3=bi2; bv2=bv1; bi2=bi1; bv1=a2; bi1=f; }
        else if (a2 > bv2) { bv3=bv2; bi3=bi2; bv2=a2; bi2=f; }
        else if (a2 > bv3) { bv3=a2; bi3=f; }
    }
    int bis[4] = {bi0, bi1, bi2, bi3};
#pragma unroll
    for (int j = 0; j < 4; ++j) {
        int f = bis[j];
        float re = Re[(size_t)f * 512];
        float im = Im[(size_t)f * 512];
        pA[idx * 4 + j]   = 2.0f * sqrtf(re * re + im * im) / 512.0f;
        pPhi[idx * 4 + j] = atan2f(im, re);
        pF[idx * 4 + j]   = f;
    }
}

// ---------------------------------------------------------------------------
// K3a: per-(b,d): x_rem = x - x_season; running cumsum -> Srem[b][t][d]
//      (Srem[t] holds S[t+1] = sum_{tau<=t} x_rem; S[0]=0 implicit).
// ---------------------------------------------------------------------------
__global__ void pass1_cumsum_kernel(const float* __restrict__ X,
                                    const float* __restrict__ pA,
                                    const float* __restrict__ pPhi,
                                    const int* __restrict__ pF,
                                    float* __restrict__ Srem) {
    int idx = blockIdx.x * blockDim.x + threadIdx.x;
    int b = idx >> 9;
    int d = idx & 511;
    float A[4], Ph[4];
    int   F[4];
#pragma unroll
    for (int j = 0; j < 4; ++j) {
        A[j]  = pA[idx * 4 + j];
        Ph[j] = pPhi[idx * 4 + j];
        F[j]  = pF[idx * 4 + j];
    }
    const float* xp = X + (size_t)b * T_ * D_ + d;
    float*       sp = Srem + (size_t)b * T_ * D_ + d;
    float run = 0.f;
    for (int t = 0; t < T_; ++t) {
        float xv = xp[(size_t)t * D_];
        float se = 0.f;
#pragma unroll
        for (int j = 0; j < 4; ++j)
            se += A[j] * cosf((float)((F[j] * t) & 511) * PI2_OVER_T + Ph[j]);
        run += xv - se;
        sp[(size_t)t * D_] = run;
    }
}

// ---------------------------------------------------------------------------
// K3b: per-(b,d): moving averages (edge-replicated) from cumsum, softmax trend
// mix, x_sum = x_rem + 2*x_season + x_trend, reduce over t with W_lin.
// ---------------------------------------------------------------------------
__global__ void pass2_trend_kernel(const float* __restrict__ Srem,
                                   const float* __restrict__ pA,
                                   const float* __restrict__ pPhi,
                                   const int* __restrict__ pF,
                                   const float* __restrict__ W_trend,
                                   const float* __restrict__ b_trend,
                                   const float* __restrict__ W_lin,
                                   const float* __restrict__ b_lin,
                                   float* __restrict__ xtrans) {
    int idx = blockIdx.x * blockDim.x + threadIdx.x;
    int b = idx >> 9;
    int d = idx & 511;
    float A[4], Ph[4];
    int   F[4];
#pragma unroll
    for (int j = 0; j < 4; ++j) {
        A[j]  = pA[idx * 4 + j];
        Ph[j] = pPhi[idx * 4 + j];
        F[j]  = pF[idx * 4 + j];
    }
    float wt[NKS], bt[NKS];
#pragma unroll
    for (int j = 0; j < NKS; ++j) { wt[j] = W_trend[j]; bt[j] = b_trend[j]; }

    const float* Sb = Srem + (size_t)b * T_ * D_ + d;
    auto Sv = [&](int i) -> float {            // S[i], i in [0,512]
        return (i <= 0) ? 0.f : Sb[(size_t)(i - 1) * D_];
    };
    const float x0 = Sv(1);                    // x_rem[0]
    const float xl = Sv(512) - Sv(511);        // x_rem[T-1]
    const int KSv[NKS] = {4, 8, 12, 16, 24, 32};

    float acc = 0.f;
    for (int t = 0; t < T_; ++t) {
        float St  = Sv(t);
        float St1 = Sv(t + 1);
        float xrem = St1 - St;

        float mm[NKS];
#pragma unroll
        for (int j = 0; j < NKS; ++j) {
            int k  = KSv[j];
            int e  = (k - 1) >> 1;
            int fr = k - 1 - e;
            int lo = t - fr;
            int hi = t + e;
            float s = 0.f;
            if (lo < 0) { s += (float)(-lo) * x0; lo = 0; }
            int hic = hi + 1;
            if (hic > 512) { s += (float)(hic - 512) * xl; hic = 512; }
            s += Sv(hic) - Sv(lo);
            mm[j] = s / (float)k;
        }
        // softmax mixture over the 6 kernel widths
        float z[NKS], mx = -3.0e38f;
#pragma unroll
        for (int j = 0; j < NKS; ++j) { z[j] = xrem * wt[j] + bt[j]; mx = fmaxf(mx, z[j]); }
        float den = 0.f, num = 0.f;
#pragma unroll
        for (int j = 0; j < NKS; ++j) {
            float ev = expf(z[j] - mx);
            den += ev;
            num += ev * mm[j];
        }
        float xtrend = num / den;

        float se = 0.f;
#pragma unroll
        for (int j = 0; j < 4; ++j)
            se += A[j] * cosf((float)((F[j] * t) & 511) * PI2_OVER_T + Ph[j]);

        float xsum = xrem + 2.f * se + xtrend;   // x + x_season + x_trend
        acc += xsum * W_lin[t];
    }
    xtrans[idx] = acc + b_lin[0];
}

// ---------------------------------------------------------------------------
// K4: routing head. One block per batch row: x_trans (512) -> base/ns (8),
// softplus, noisy logits, softmax, top-4 scatter into out[b][8].
// ---------------------------------------------------------------------------
__global__ __launch_bounds__(256) void head_kernel(
        const float* __restrict__ xtrans,
        const float* __restrict__ noise,
        const float* __restrict__ W_r,  const float* __restrict__ b_r,
        const float* __restrict__ W_n,  const float* __restrict__ b_n,
        float* __restrict__ out) {
    __shared__ float sB[256 * M_];
    __shared__ float sN[256 * M_];
    const int b   = blockIdx.x;
    const int tid = threadIdx.x;

    float aB[M_], aN[M_];
#pragma unroll
    for (int m = 0; m < M_; ++m) { aB[m] = 0.f; aN[m] = 0.f; }
    for (int d = tid; d < D_; d += 256) {
        float xt = xtrans[b * D_ + d];
#pragma unroll
        for (int m = 0; m < M_; ++m) {
            aB[m] += xt * W_r[d * M_ + m];
            aN[m] += xt * W_n[d * M_ + m];
        }
    }
#pragma unroll
    for (int m = 0; m < M_; ++m) { sB[tid * M_ + m] = aB[m]; sN[tid * M_ + m] = aN[m]; }
    __syncthreads();
    for (int s = 128; s > 0; s >>= 1) {
        if (tid < s) {
#pragma unroll
            for (int m = 0; m < M_; ++m) {
                sB[tid * M_ + m] += sB[(tid + s) * M_ + m];
                sN[tid * M_ + m] += sN[(tid + s) * M_ + m];
            }
        }
        __syncthreads();
    }
    if (tid == 0) {
        float g[M_];
        float mx = -3.0e38f;
#pragma unroll
        for (int m = 0; m < M_; ++m) {
            float base = sB[m] + b_r[m];
            float zn   = sN[m] + b_n[m];
            // stable softplus = max(z,0) + log1p(exp(-|z|))
            float sp = fmaxf(zn, 0.f) + log1pf(expf(-fabsf(zn)));
            g[m] = base + noise[b * M_ + m] * sp;
            mx = fmaxf(mx, g[m]);
        }
        float den = 0.f;
        float pw[M_];
#pragma unroll
        for (int m = 0; m < M_; ++m) { pw[m] = expf(g[m] - mx); den += pw[m]; }
#pragma unroll
        for (int m = 0; m < M_; ++m) pw[m] /= den;

        // top-4 (stable, lowest index on ties)
        float bv[KSEL] = {-1.f, -1.f, -1.f, -1.f};
        int   bi[KSEL] = {0, 0, 0, 0};
        for (int m = 0; m < M_; ++m) {
            float v = pw[m];
            if (v > bv[0])      { bv[3]=bv[2]; bi[3]=bi[2]; bv[2]=bv[1]; bi[2]=bi[1]; bv[1]=bv[0]; bi[1]=bi[0]; bv[0]=v; bi[0]=m; }
            else if (v > bv[1]) { bv[3]=bv[2]; bi[3]=bi[2]; bv[2]=bv[1]; bi[2]=bi[1]; bv[1]=v; bi[1]=m; }
            else if (v > bv[2]) { bv[3]=bv[2]; bi[3]=bi[2]; bv[2]=v; bi[2]=m; }
            else if (v > bv[3]) { bv[3]=v; bi[3]=m; }
        }
        float o[M_];
#pragma unroll
        for (int m = 0; m < M_; ++m) o[m] = 0.f;
#pragma unroll
        for (int j = 0; j < KSEL; ++j) o[bi[j]] = bv[j];
#pragma unroll
        for (int m = 0; m < M_; ++m) out[b * M_ + m] = o[m];
    }
}

// ---------------------------------------------------------------------------
extern "C" void kernel_launch(void* const* d_in, const int* in_sizes, int n_in,
                              void* d_out, int out_size, void* d_ws, size_t ws_size,
                              hipStream_t stream) {
    (void)in_sizes; (void)n_in; (void)out_size; (void)ws_size;
    const float* x       = (const float*)d_in[0];   // (B,T,D,1)
    const float* noise   = (const float*)d_in[1];   // (B,M)
    const float* W_r     = (const float*)d_in[2];   // (D,M)
    const float* b_r     = (const float*)d_in[3];   // (M,)
    const float* W_noise = (const float*)d_in[4];   // (D,M)
    const float* b_noise = (const float*)d_in[5];   // (M,)
    const float* W_trend = (const float*)d_in[6];   // (6,)
    const float* b_trend = (const float*)d_in[7];   // (6,)
    const float* W_lin   = (const float*)d_in[8];   // (T,)
    const float* b_lin   = (const float*)d_in[9];   // (1,)
    float* out = (float*)d_out;                     // (B,M)

    // workspace carve (floats): ~35 MB total
    float* ws     = (float*)d_ws;
    float* Wall   = ws;                               // 512*512
    float* Xall   = Wall + 512 * 512;                 // B*512*512
    float* Srem   = Xall + (size_t)B_ * 512 * 512;    // B*T*D
    float* pA     = Srem + (size_t)B_ * T_ * D_;      // B*D*4
    float* pPhi   = pA + (size_t)B_ * D_ * 4;         // B*D*4
    int*   pF     = (int*)(pPhi + (size_t)B_ * D_ * 4);
    float* xtrans = (float*)(pF + (size_t)B_ * D_ * 4);  // B*D

    build_coeffs_kernel<<<(512 * 512) / 256, 256, 0, stream>>>(Wall);
    dft_gemm_kernel<<<dim3(16, B_), 128, 0, stream>>>(Wall, x, Xall);
    select_topk_kernel<<<(B_ * D_) / 256, 256, 0, stream>>>(Xall, pA, pPhi, pF);
    pass1_cumsum_kernel<<<(B_ * D_) / 256, 256, 0, stream>>>(x, pA, pPhi, pF, Srem);
    pass2_trend_kernel<<<(B_ * D_) / 256, 256, 0, stream>>>(Srem, pA, pPhi, pF,
                                                            W_trend, b_trend, W_lin, b_lin,
                                                            xtrans);
    head_kernel<<<B_, 256, 0, stream>>>(xtrans, noise, W_r, b_r, W_noise, b_noise, out);
}